// MaskModel_3934190043281
// MI455X (gfx1250) — compile-verified
//
#include <hip/hip_runtime.h>
#include <hip/hip_bf16.h>

typedef __attribute__((ext_vector_type(2))) float v2f;
typedef __attribute__((ext_vector_type(8))) float v8f;

#define S_LEN 12
#define D_EMB 10
#define BLK 256
#define WPB 8   // waves per block (wave32)

// ---- shared-weight LDS offsets (floats) ----
// 0:We_w(10) 10:We_b(10) 20:Wt_w(10) 30:Wt_b(10)
// 40:W1_w(1000) 1040:W1_b(50) 1090:W2_w(500) 1590:W2_b(10)
// 1600:W3_w(10) 1610:W3_b(1)  total 1611
#define SW_TOTAL 1611
#define XS_STRIDE 21   // 20 cols + pad (odd stride -> bank-conflict free)
#define HS_STRIDE 51   // 50 cols + pad

__device__ __forceinline__ void cpy_lds(float* dst, const float* src, int n, int tid) {
    for (int i = tid; i < n; i += BLK) dst[i] = src[i];
}

// Per-sample attention front-end (algebraically reduced; matches reference math).
__device__ __forceinline__ void frontend(const float* __restrict__ inp, float t,
                                         const float* Wew, const float* Web,
                                         const float* Wtw, const float* Wtb,
                                         float* embt /*[10]*/, float* attn /*[12]*/,
                                         float* asum_out, float* wsum_out) {
    float a = 0.f, c = 0.f;
#pragma unroll
    for (int d = 0; d < D_EMB; ++d) {
        float et = fmaf(t, Wtw[d], Wtb[d]);
        embt[d] = et;
        a = fmaf(Wew[d], et, a);
        c = fmaf(Web[d], et, c);
    }
    float in[S_LEN];
    const float4* p4 = (const float4*)inp;  // b*12 floats -> 48B aligned
    float4 v0 = p4[0], v1 = p4[1], v2 = p4[2];
    in[0]=v0.x; in[1]=v0.y; in[2]=v0.z; in[3]=v0.w;
    in[4]=v1.x; in[5]=v1.y; in[6]=v1.z; in[7]=v1.w;
    in[8]=v2.x; in[9]=v2.y; in[10]=v2.z; in[11]=v2.w;

    float sc[S_LEN];
    float m = -3.402823466e+38f;
#pragma unroll
    for (int s = 0; s < S_LEN; ++s) { sc[s] = fmaf(in[s], a, c); m = fmaxf(m, sc[s]); }
    float zmax = -3.402823466e+38f;
#pragma unroll
    for (int s = 0; s < S_LEN; ++s) { sc[s] = sc[s] / m; zmax = fmaxf(zmax, sc[s]); }
    float esum = 0.f;
#pragma unroll
    for (int s = 0; s < S_LEN; ++s) { float e = expf(sc[s] - zmax); attn[s] = e; esum += e; }
    float inv = 1.f / esum;
    float asum = 0.f, wsum = 0.f;
#pragma unroll
    for (int s = 0; s < S_LEN; ++s) {
        attn[s] *= inv;
        asum += attn[s];
        wsum = fmaf(attn[s], in[s], wsum);
    }
    *asum_out = asum;
    *wsum_out = wsum;
}

// ===================== Kernel 1: fused forward (WMMA MLP) =====================
__global__ __launch_bounds__(BLK) void fused_fwd(
    const float* __restrict__ inputs, const float* __restrict__ targets,
    const float* __restrict__ We_w, const float* __restrict__ We_b,
    const float* __restrict__ Wt_w, const float* __restrict__ Wt_b,
    const float* __restrict__ W1_w, const float* __restrict__ W1_b,
    const float* __restrict__ W2_w, const float* __restrict__ W2_b,
    const float* __restrict__ W3_w, const float* __restrict__ W3_b,
    float* __restrict__ out_tm, float* __restrict__ blk_partials) {
    __shared__ float sW[SW_TOTAL];
    __shared__ float sX[WPB][32 * XS_STRIDE];   // x[20] then reused for h2[10]
    __shared__ float sH[WPB][32 * HS_STRIDE];   // h1[50]
    __shared__ float sRed[WPB];

    const int tid = threadIdx.x;
    cpy_lds(sW + 0,    We_w, 10, tid);
    cpy_lds(sW + 10,   We_b, 10, tid);
    cpy_lds(sW + 20,   Wt_w, 10, tid);
    cpy_lds(sW + 30,   Wt_b, 10, tid);
    cpy_lds(sW + 40,   W1_w, 1000, tid);
    cpy_lds(sW + 1040, W1_b, 50, tid);
    cpy_lds(sW + 1090, W2_w, 500, tid);
    cpy_lds(sW + 1590, W2_b, 10, tid);
    cpy_lds(sW + 1600, W3_w, 10, tid);
    cpy_lds(sW + 1610, W3_b, 1, tid);
    __syncthreads();

    const int wave = tid >> 5;
    const int lane = tid & 31;
    const int half = lane >> 4;   // 0: lanes 0-15, 1: lanes 16-31
    const int lm   = lane & 15;
    const int b = blockIdx.x * BLK + tid;

    // ---- front-end: attention -> x = [emb_t | ctx] ----
    float embt[D_EMB], attn[S_LEN], asum, wsum;
    frontend(inputs + (size_t)b * S_LEN, targets[b],
             sW + 0, sW + 10, sW + 20, sW + 30, embt, attn, &asum, &wsum);

    float* xsw = sX[wave];
    float* xrow = xsw + lane * XS_STRIDE;
#pragma unroll
    for (int d = 0; d < D_EMB; ++d) xrow[d] = embt[d];
#pragma unroll
    for (int d = 0; d < D_EMB; ++d)
        xrow[D_EMB + d] = fmaf(sW[d], wsum, sW[10 + d] * asum);  // ctx[d]

    // deterministic block partial-sum of attn (for global mean)
    float r = asum;
#pragma unroll
    for (int off = 16; off; off >>= 1) r += __shfl_down(r, off, 32);
    if (lane == 0) sRed[wave] = r;
    __syncthreads();   // also publishes sX for WMMA phase
    if (tid == 0) {
        float t2 = 0.f;
#pragma unroll
        for (int w = 0; w < WPB; ++w) t2 += sRed[w];
        blk_partials[blockIdx.x] = t2;
    }

    // ---- Layer 1: [32,20] x [20,50] via V_WMMA_F32_16X16X4_F32 ----
    const float* sW1  = sW + 40;
    const float* sW1B = sW + 1040;
    float* hsw = sH[wave];

    // A tiles: A[m][k], lane half picks k0/k0+1 pair (16x4 f32 layout)
    v2f a1[2][5];
#pragma unroll
    for (int mt = 0; mt < 2; ++mt)
#pragma unroll
        for (int kk = 0; kk < 5; ++kk) {
            int row = mt * 16 + lm;
            int k0 = kk * 4 + half * 2;
            a1[mt][kk].x = xsw[row * XS_STRIDE + k0];
            a1[mt][kk].y = xsw[row * XS_STRIDE + k0 + 1];
        }

#pragma unroll
    for (int nt = 0; nt < 4; ++nt) {
        int col = nt * 16 + lm;
        bool cv = col < 50;
        v2f bm[5];
#pragma unroll
        for (int kk = 0; kk < 5; ++kk) {
            int k0 = kk * 4 + half * 2;
            bm[kk].x = cv ? sW1[col * 20 + k0]     : 0.f;   // B[k][n] = W1[n][k]
            bm[kk].y = cv ? sW1[col * 20 + k0 + 1] : 0.f;
        }
        float bias = cv ? sW1B[col] : 0.f;
#pragma unroll
        for (int mt = 0; mt < 2; ++mt) {
            v8f acc = {0.f, 0.f, 0.f, 0.f, 0.f, 0.f, 0.f, 0.f};
#pragma unroll
            for (int kk = 0; kk < 5; ++kk)
                acc = __builtin_amdgcn_wmma_f32_16x16x4_f32(
                    false, a1[mt][kk], false, bm[kk], (short)0, acc, false, false);
            if (cv) {
#pragma unroll
                for (int v = 0; v < 8; ++v) {
                    int row = mt * 16 + v + half * 8;
                    hsw[row * HS_STRIDE + col] = fmaxf(acc[v] + bias, 0.f);
                }
            }
        }
    }
    __syncthreads();

    // ---- Layer 2: [32,50] x [50,10] ----
    const float* sW2  = sW + 1090;
    const float* sW2B = sW + 1590;
    bool nv = lm < 10;
    v2f b2m[13];
#pragma unroll
    for (int kk = 0; kk < 13; ++kk) {
        int k0 = kk * 4 + half * 2;
        b2m[kk].x = (nv && k0     < 50) ? sW2[lm * 50 + k0]     : 0.f;
        b2m[kk].y = (nv && k0 + 1 < 50) ? sW2[lm * 50 + k0 + 1] : 0.f;
    }
    float bias2 = nv ? sW2B[lm] : 0.f;
#pragma unroll
    for (int mt = 0; mt < 2; ++mt) {
        v8f acc = {0.f, 0.f, 0.f, 0.f, 0.f, 0.f, 0.f, 0.f};
#pragma unroll
        for (int kk = 0; kk < 13; ++kk) {
            int k0 = kk * 4 + half * 2;
            int row = mt * 16 + lm;
            v2f am;
            am.x = (k0     < 50) ? hsw[row * HS_STRIDE + k0]     : 0.f;
            am.y = (k0 + 1 < 50) ? hsw[row * HS_STRIDE + k0 + 1] : 0.f;
            acc = __builtin_amdgcn_wmma_f32_16x16x4_f32(
                false, am, false, b2m[kk], (short)0, acc, false, false);
        }
        if (nv) {
#pragma unroll
            for (int v = 0; v < 8; ++v) {
                int row = mt * 16 + v + half * 8;
                xsw[row * XS_STRIDE + lm] = fmaxf(acc[v] + bias2, 0.f);  // h2
            }
        }
    }
    __syncthreads();

    // ---- Layer 3: 10-MAC dot per sample ----
    const float* sW3 = sW + 1600;
    float tm = sW[1610];
    const float* h2row = xsw + lane * XS_STRIDE;
#pragma unroll
    for (int n = 0; n < D_EMB; ++n) tm = fmaf(h2row[n], sW3[n], tm);
    out_tm[b] = tm;
}

// ===================== Kernel 2: reduce partials -> mean =====================
__global__ __launch_bounds__(BLK) void reduce_mean(const float* __restrict__ partials,
                                                   float* __restrict__ mean_out,
                                                   int nblk, float inv_count) {
    __shared__ float red[WPB];
    int tid = threadIdx.x;
    float s = 0.f;
    for (int i = tid; i < nblk; i += BLK) s += partials[i];
#pragma unroll
    for (int off = 16; off; off >>= 1) s += __shfl_down(s, off, 32);
    int wave = tid >> 5, lane = tid & 31;
    if (lane == 0) red[wave] = s;
    __syncthreads();
    if (tid == 0) {
        float t = 0.f;
#pragma unroll
        for (int w = 0; w < WPB; ++w) t += red[w];
        *mean_out = t * inv_count;
    }
}

// ===================== Kernel 3: mask = attn > mean =====================
__global__ __launch_bounds__(BLK) void mask_kernel(
    const float* __restrict__ inputs, const float* __restrict__ targets,
    const float* __restrict__ We_w, const float* __restrict__ We_b,
    const float* __restrict__ Wt_w, const float* __restrict__ Wt_b,
    const float* __restrict__ mean_ptr, float* __restrict__ out_mask) {
    int b = blockIdx.x * BLK + threadIdx.x;
    float embt[D_EMB], attn[S_LEN], asum, wsum;
    frontend(inputs + (size_t)b * S_LEN, targets[b], We_w, We_b, Wt_w, Wt_b,
             embt, attn, &asum, &wsum);
    float mean = *mean_ptr;
    float v[S_LEN];
#pragma unroll
    for (int s = 0; s < S_LEN; ++s) v[s] = (attn[s] > mean) ? 1.f : 0.f;
    float4* d4 = (float4*)(out_mask + (size_t)b * S_LEN);
    d4[0] = make_float4(v[0], v[1], v[2], v[3]);
    d4[1] = make_float4(v[4], v[5], v[6], v[7]);
    d4[2] = make_float4(v[8], v[9], v[10], v[11]);
}

extern "C" void kernel_launch(void* const* d_in, const int* in_sizes, int n_in,
                              void* d_out, int out_size, void* d_ws, size_t ws_size,
                              hipStream_t stream) {
    const float* inputs  = (const float*)d_in[0];
    const float* targets = (const float*)d_in[1];
    const float* We_w = (const float*)d_in[2];
    const float* We_b = (const float*)d_in[3];
    const float* Wt_w = (const float*)d_in[4];
    const float* Wt_b = (const float*)d_in[5];
    const float* W1_w = (const float*)d_in[6];
    const float* W1_b = (const float*)d_in[7];
    const float* W2_w = (const float*)d_in[8];
    const float* W2_b = (const float*)d_in[9];
    const float* W3_w = (const float*)d_in[10];
    const float* W3_b = (const float*)d_in[11];

    const int B = in_sizes[1];          // targets is [B,1]
    const int nblk = B / BLK;           // B = 524288 -> 2048 full blocks

    float* out_tm   = (float*)d_out;            // [B]
    float* out_mask = (float*)d_out + B;        // [B*S] as 0.0/1.0

    float* partials = (float*)d_ws;             // [nblk]
    float* mean_p   = partials + nblk;          // [1]

    fused_fwd<<<nblk, BLK, 0, stream>>>(inputs, targets, We_w, We_b, Wt_w, Wt_b,
                                        W1_w, W1_b, W2_w, W2_b, W3_w, W3_b,
                                        out_tm, partials);
    reduce_mean<<<1, BLK, 0, stream>>>(partials, mean_p, nblk,
                                       1.0f / ((float)B * (float)S_LEN));
    mask_kernel<<<nblk, BLK, 0, stream>>>(inputs, targets, We_w, We_b, Wt_w, Wt_b,
                                          mean_p, out_mask);
}